// MissModel_79869211837047
// MI455X (gfx1250) — compile-verified
//
#include <hip/hip_runtime.h>

// Folded 20x Linear(1,1) chain: y = a*x + c, streamed at HBM rate.
// Pure bandwidth-bound (256 MiB @ 23.3 TB/s ~= 11.5 us floor); WMMA/TDM
// intentionally unused -- no matrix structure survives the fold.

typedef float f32x4 __attribute__((ext_vector_type(4)));

__global__ __launch_bounds__(256) void affine_stream_kernel(
    const float* __restrict__ x,
    const float* __restrict__ W,
    const float* __restrict__ Bv,
    float* __restrict__ y,
    int n4, int nLayers, int nTail) {
  // Fold the layer chain into (a, c). W/B reads are wave-uniform -> the
  // compiler emits s_load scalar reads (constant cache); cost is ~nothing
  // amortized over the 16 tokens each thread streams.
  float a = 1.0f;
  float c = 0.0f;
#pragma unroll 1
  for (int i = 0; i < nLayers; ++i) {
    float w = W[i];
    float b = Bv[i];
    a = a * w;
    c = __builtin_fmaf(c, w, b);
  }

  const f32x4* __restrict__ x4 = (const f32x4*)x;
  f32x4* __restrict__ y4 = (f32x4*)y;

  const int tid    = blockIdx.x * blockDim.x + threadIdx.x;
  const int stride = gridDim.x * blockDim.x;

  // 4-way unrolled grid-stride: 4 non-temporal B128 loads in flight per
  // lane (512B/lane-group per step, fully coalesced across the wave32).
  int i = tid;
  for (; i + 3 * stride < n4; i += 4 * stride) {
    f32x4 v0 = __builtin_nontemporal_load(&x4[i]);
    f32x4 v1 = __builtin_nontemporal_load(&x4[i + stride]);
    f32x4 v2 = __builtin_nontemporal_load(&x4[i + 2 * stride]);
    f32x4 v3 = __builtin_nontemporal_load(&x4[i + 3 * stride]);
    __builtin_nontemporal_store(v0 * a + c, &y4[i]);
    __builtin_nontemporal_store(v1 * a + c, &y4[i + stride]);
    __builtin_nontemporal_store(v2 * a + c, &y4[i + 2 * stride]);
    __builtin_nontemporal_store(v3 * a + c, &y4[i + 3 * stride]);
  }
  // Remainder vec4s (none with the exact launch below, kept for safety).
  for (; i < n4; i += stride) {
    f32x4 v = __builtin_nontemporal_load(&x4[i]);
    __builtin_nontemporal_store(v * a + c, &y4[i]);
  }
  // Scalar tail for n % 4 (zero for N = 2^25).
  if (tid == 0 && nTail) {
    const int base = n4 * 4;
    for (int k = 0; k < nTail; ++k) {
      y[base + k] = __builtin_fmaf(a, x[base + k], c);
    }
  }
}

extern "C" void kernel_launch(void* const* d_in, const int* in_sizes, int n_in,
                              void* d_out, int out_size, void* d_ws, size_t ws_size,
                              hipStream_t stream) {
  (void)n_in; (void)out_size; (void)d_ws; (void)ws_size;
  const float* x  = (const float*)d_in[0];  // [N, 1] f32
  const float* W  = (const float*)d_in[1];  // [L, 1, 1] f32
  const float* Bv = (const float*)d_in[2];  // [L, 1] f32
  float* y = (float*)d_out;

  const int n  = in_sizes[0];   // 2^25
  const int L  = in_sizes[1];   // 20
  const int n4 = n >> 2;
  const int tail = n & 3;

  const int threads = 256;                 // 8 wave32 per block
  // Each thread handles 4 vec4 (64 B in, 64 B out) -> exact coverage.
  long long thread_goal = ((long long)n4 + 3) / 4;
  int blocks = (int)((thread_goal + threads - 1) / threads);
  if (blocks < 1) blocks = 1;

  affine_stream_kernel<<<blocks, threads, 0, stream>>>(x, W, Bv, y, n4, L, tail);
}